// Ranking_Loss_71382356459609
// MI455X (gfx1250) — compile-verified
//
#include <hip/hip_runtime.h>
#include <hip/hip_bf16.h>
#include <math.h>

typedef float v2f __attribute__((ext_vector_type(2)));
typedef float v8f __attribute__((ext_vector_type(8)));

#define ONE_PLUS_SIGMA 1.15f

// Full wave32 sum using V_WMMA_F32_16X16X4_F32 with B = ones.
// A layout (32-bit 16x4): lanes 0-15 VGPR0 -> A[m][0], lanes 16-31 VGPR0 -> A[m][2].
// With a[1]=0: D[m][n] = x_m + x_{m+16}, constant across n.
// C/D layout: lane L<16 VGPR v = D[v][L]; lane L>=16 VGPR v = D[v+8][L-16].
// => per-lane sum of 8 C VGPRs = rows 0-7 (lanes 0-15) or rows 8-15 (lanes 16-31);
// one xor-16 shuffle gives the full 32-lane total in every lane.
__device__ __forceinline__ float wave_sum_wmma(float x) {
    v2f a; a[0] = x;    a[1] = 0.0f;
    v2f b; b[0] = 1.0f; b[1] = 1.0f;
    v8f c = {0.0f, 0.0f, 0.0f, 0.0f, 0.0f, 0.0f, 0.0f, 0.0f};
    c = __builtin_amdgcn_wmma_f32_16x16x4_f32(false, a, false, b, (short)0, c, false, false);
    float s = ((c[0] + c[1]) + (c[2] + c[3])) + ((c[4] + c[5]) + (c[6] + c[7]));
    s += __shfl_xor(s, 16, 32);
    return s;
}

// Pass 1: per-pair math + in-block reduction; one deterministic partial
// (lg, sq, n1, n0) per block written to the workspace.
__global__ __launch_bounds__(256) void pair_partial_kernel(
    const float* __restrict__ p, const float* __restrict__ g,
    const int* __restrict__ ia, const int* __restrict__ ib,
    int n, float* __restrict__ partials) {

    float lg = 0.0f, sq = 0.0f, c1 = 0.0f, c0 = 0.0f;

    const int tid    = blockIdx.x * blockDim.x + threadIdx.x;
    const int stride = gridDim.x * blockDim.x;

#define PROC(ai, bi) do {                                                    \
        const int a_ = (ai), b_ = (bi);                                      \
        const float za = g[a_], zb = g[b_];                                  \
        const float t = (za / zb >= ONE_PLUS_SIGMA) ? 1.0f                   \
                        : ((zb / za > ONE_PLUS_SIGMA) ? -1.0f : 0.0f);       \
        const float d = p[a_] - p[b_];                                       \
        const float x = -t * d;                                              \
        const float sp = fmaxf(x, 0.0f) + log1pf(__expf(-fabsf(x)));         \
        if (t != 0.0f) { lg += sp;    c1 += 1.0f; }                          \
        else           { sq += d * d; c0 += 1.0f; }                          \
    } while (0)

    // Vectorized b128 index loads (both index arrays are 16B-aligned bases).
    const int  n4  = n >> 2;
    const int4* ia4 = (const int4*)ia;
    const int4* ib4 = (const int4*)ib;
    for (int i = tid; i < n4; i += stride) {
        __builtin_prefetch(&ia4[i + stride], 0, 1);   // -> global_prefetch_b8
        __builtin_prefetch(&ib4[i + stride], 0, 1);
        const int4 A = ia4[i];
        const int4 B = ib4[i];
        PROC(A.x, B.x); PROC(A.y, B.y); PROC(A.z, B.z); PROC(A.w, B.w);
    }
    for (int i = (n4 << 2) + tid; i < n; i += stride) PROC(ia[i], ib[i]);
#undef PROC

    // Wave reduction on the matrix pipe (EXEC all-1s here: no early exits).
    const float wlg = wave_sum_wmma(lg);
    const float wsq = wave_sum_wmma(sq);
    const float wc1 = wave_sum_wmma(c1);
    const float wc0 = wave_sum_wmma(c0);

    __shared__ float sacc[4][8];
    const int wid  = threadIdx.x >> 5;   // wave32: 8 waves per 256-thread block
    const int lane = threadIdx.x & 31;
    if (lane == 0) {
        sacc[0][wid] = wlg; sacc[1][wid] = wsq;
        sacc[2][wid] = wc1; sacc[3][wid] = wc0;
    }
    __syncthreads();
    if (threadIdx.x < 4) {
        float v = 0.0f;
        const int nw = blockDim.x >> 5;
        for (int w = 0; w < nw; ++w) v += sacc[threadIdx.x][w];
        partials[blockIdx.x * 4 + threadIdx.x] = v;
    }
}

// Pass 2: fixed-order (deterministic) reduction of all block partials + epilogue.
__global__ __launch_bounds__(256) void finalize_kernel(
    const float* __restrict__ partials, int nslots, float* __restrict__ out) {
    __shared__ float s[4][256];
    float a0 = 0.0f, a1 = 0.0f, a2 = 0.0f, a3 = 0.0f;
    for (int i = threadIdx.x; i < nslots; i += blockDim.x) {
        a0 += partials[i * 4 + 0];
        a1 += partials[i * 4 + 1];
        a2 += partials[i * 4 + 2];
        a3 += partials[i * 4 + 3];
    }
    s[0][threadIdx.x] = a0; s[1][threadIdx.x] = a1;
    s[2][threadIdx.x] = a2; s[3][threadIdx.x] = a3;
    __syncthreads();
    for (int off = 128; off > 0; off >>= 1) {
        if ((int)threadIdx.x < off) {
            s[0][threadIdx.x] += s[0][threadIdx.x + off];
            s[1][threadIdx.x] += s[1][threadIdx.x + off];
            s[2][threadIdx.x] += s[2][threadIdx.x + off];
            s[3][threadIdx.x] += s[3][threadIdx.x + off];
        }
        __syncthreads();
    }
    if (threadIdx.x == 0) {
        const float lg_sum = s[0][0], sq_sum = s[1][0];
        const float n1 = s[2][0], n0 = s[3][0];
        const float lg = lg_sum / fmaxf(n1, 1.0f);
        const float sq = sq_sum / fmaxf(n0, 1.0f);
        out[0] = (n1 == 0.0f) ? sq : ((n0 == 0.0f) ? lg : (lg + sq));
    }
}

extern "C" void kernel_launch(void* const* d_in, const int* in_sizes, int n_in,
                              void* d_out, int out_size, void* d_ws, size_t ws_size,
                              hipStream_t stream) {
    const float* p  = (const float*)d_in[0];  // pred_depth (flat)
    const float* g  = (const float*)d_in[1];  // gt_depth (flat)
    const int*   ra = (const int*)d_in[2];    // idx_rand_a
    const int*   rb = (const int*)d_in[3];    // idx_rand_b
    const int*   oa = (const int*)d_in[4];    // idx_obj_inval
    const int*   ob = (const int*)d_in[5];    // idx_obj_val
    const int k1 = in_sizes[2];
    const int k2 = in_sizes[4];

    // Blocks per pair-set; shrink if workspace is tiny (constant across calls
    // -> still deterministic). ws layout: [2*nblk][4] floats of partials.
    int nblk = 512;
    while (nblk > 1 && (size_t)(2 * nblk * 4) * sizeof(float) > ws_size) nblk >>= 1;

    float* part = (float*)d_ws;
    pair_partial_kernel<<<nblk, 256, 0, stream>>>(p, g, ra, rb, k1, part);
    pair_partial_kernel<<<nblk, 256, 0, stream>>>(p, g, oa, ob, k2, part + (size_t)nblk * 4);
    finalize_kernel<<<1, 256, 0, stream>>>(part, 2 * nblk, (float*)d_out);
}